// SiSLoss_5583457485018
// MI455X (gfx1250) — compile-verified
//
#include <hip/hip_runtime.h>
#include <hip/hip_bf16.h>
#include <stdint.h>

// SiSLoss: windowed shift-invariant L1, N=4 C=4 H=W=384, 9x9 shifts, stride 4.
// cost(n,s,h,w) = sum_c |ms[n,c,h,w] - target_reflectpad_shift_s[n,c,h,w]|
// out = mean over (n,h,w) of min over s.

#define H_DIM 384
#define W_DIM 384
#define N_DIM 4
#define C_DIM 4
#define TS 32            // output tile side (32x32 pixels per block)
#define HALO 16          // ns * stride = 4*4
#define TP (TS + 2*HALO) // 64: padded tile side (power of two -> shift/mask addressing)
#define TILE_PIX (TP*TP) // 4096
#define NBLOCKS (N_DIM * (H_DIM/TS) * (W_DIM/TS)) // 4 * 12 * 12 = 576

__device__ __forceinline__ int reflect_idx(int v) {
    // jnp.pad(..., mode="reflect"): -k -> k ; (H-1)+k -> (H-1)-k  (pad=16 < H, single bounce)
    if (v < 0)       v = -v;
    if (v >= H_DIM)  v = 2*H_DIM - 2 - v;
    return v;
}

__global__ __launch_bounds__(256)
void sis_main_kernel(const float* __restrict__ ms_img,
                     const float* __restrict__ target,
                     float* __restrict__ partial) {
    // Single shared object -> guaranteed LDS offset 0 for the async stores.
    // tile: channel-interleaved padded target tile, one float4 per pixel (64 KB).
    // After the shift loop the tile is dead; its first 64 float4s are reused
    // as the 256-float block-reduction scratch.
    __shared__ float4 tile[TILE_PIX];

    const int tid    = threadIdx.x;
    const int n      = blockIdx.z;
    const int tileY0 = blockIdx.y * TS;
    const int tileX0 = blockIdx.x * TS;

    // ---- Stage 64x64x4 reflect-padded target tile into LDS via async gathers ----
    // Per channel: 4096 pixels / 256 threads = 16 passes, coalesced along x
    // except at reflection seams. LDS float layout: ((y*64+x)*4 + c).
    #pragma unroll
    for (int c = 0; c < C_DIM; ++c) {
        const float* tgtc = target + (size_t)(n * C_DIM + c) * (H_DIM * W_DIM);
        #pragma unroll
        for (int k = 0; k < TILE_PIX / 256; ++k) {
            int pix = tid + k * 256;          // 0..4095
            int y   = pix >> 6;
            int x   = pix & 63;
            int gy  = reflect_idx(tileY0 - HALO + y);
            int gx  = reflect_idx(tileX0 - HALO + x);
            unsigned long long gaddr =
                (unsigned long long)(tgtc + (size_t)gy * W_DIM + gx);
            unsigned ldsoff = (unsigned)((pix << 4) + (c << 2));
            // CDNA5 async global->LDS gather (ASYNCcnt), no VGPR round-trip.
            asm volatile("global_load_async_to_lds_b32 %0, %1, off"
                         :: "v"(ldsoff), "v"(gaddr)
                         : "memory");
        }
    }
    asm volatile("s_wait_asynccnt 0" ::: "memory");
    __syncthreads();

    // ---- 4 pixels per thread: 4 independent abs/add/min chains (ILP) ----
    const size_t ms_base = (size_t)n * C_DIM * H_DIM * W_DIM;
    float acc = 0.0f;   // sum of this thread's per-pixel mins

    #pragma unroll
    for (int p = 0; p < 4; ++p) {
        const int idx = tid + p * 256;        // 0..1023 within 32x32 tile
        const int ty  = idx >> 5;             // 0..31
        const int tx  = idx & 31;             // 0..31
        const int gy  = tileY0 + ty;
        const int gx  = tileX0 + tx;

        const float* msp = ms_img + ms_base + (size_t)gy * W_DIM + gx;
        const float m0 = msp[0 * H_DIM * W_DIM];
        const float m1 = msp[1 * H_DIM * W_DIM];
        const float m2 = msp[2 * H_DIM * W_DIM];
        const float m3 = msp[3 * H_DIM * W_DIM];

        // Output pixel at padded tile coord (ty+16, tx+16); shift (i,j) samples
        // (ty + i*4, tx + j*4), i,j in 0..8 -> range [0,64).
        float minc = 3.402823466e+38f;
        #pragma unroll
        for (int i = 0; i < 9; ++i) {
            const float4* rowp = &tile[((ty + i * 4) << 6) + tx];
            #pragma unroll
            for (int j = 0; j < 9; ++j) {
                float4 t = rowp[j * 4];                   // ds_load_b128
                float cst = fabsf(m0 - t.x) + fabsf(m1 - t.y)
                          + fabsf(m2 - t.z) + fabsf(m3 - t.w);
                minc = fminf(minc, cst);
            }
        }
        acc += minc;
    }

    // ---- Deterministic block sum (tile LDS is dead; reuse its head) ----
    __syncthreads();                     // all tile reads done before reuse
    float* red = (float*)&tile[0];       // 256 floats
    red[tid] = acc;
    __syncthreads();
    #pragma unroll
    for (int s = 128; s > 0; s >>= 1) {
        if (tid < s) red[tid] += red[tid + s];
        __syncthreads();
    }
    if (tid == 0) {
        int bid = (blockIdx.z * gridDim.y + blockIdx.y) * gridDim.x + blockIdx.x;
        partial[bid] = red[0];
    }
}

__global__ __launch_bounds__(256)
void sis_finalize_kernel(const float* __restrict__ partial,
                         float* __restrict__ out) {
    __shared__ float red[256];
    float s = 0.0f;
    for (int i = threadIdx.x; i < NBLOCKS; i += 256) s += partial[i];  // fixed order
    red[threadIdx.x] = s;
    __syncthreads();
    #pragma unroll
    for (int k = 128; k > 0; k >>= 1) {
        if (threadIdx.x < k) red[threadIdx.x] += red[threadIdx.x + k];
        __syncthreads();
    }
    if (threadIdx.x == 0)
        out[0] = red[0] * (1.0f / (float)(N_DIM * H_DIM * W_DIM));
}

extern "C" void kernel_launch(void* const* d_in, const int* in_sizes, int n_in,
                              void* d_out, int out_size, void* d_ws, size_t ws_size,
                              hipStream_t stream) {
    (void)in_sizes; (void)n_in; (void)out_size; (void)ws_size;
    const float* ms_img = (const float*)d_in[0];
    const float* target = (const float*)d_in[1];
    float* out     = (float*)d_out;
    float* partial = (float*)d_ws;   // NBLOCKS floats = 2304 B of scratch

    dim3 grid(W_DIM / TS, H_DIM / TS, N_DIM);   // 12 x 12 x 4
    dim3 block(256);
    sis_main_kernel<<<grid, block, 0, stream>>>(ms_img, target, partial);
    sis_finalize_kernel<<<1, 256, 0, stream>>>(partial, out);
}